// TunableDirGNN_30339648979105
// MI455X (gfx1250) — compile-verified
//
#include <hip/hip_runtime.h>
#include <math.h>

// ---------------------------------------------------------------------------
// PNA GNN for MI455X (gfx1250, wave32). All GEMMs via V_WMMA_F32_16X16X4_F32.
// ---------------------------------------------------------------------------

typedef float v2f __attribute__((ext_vector_type(2)));
typedef float v8f __attribute__((ext_vector_type(8)));

#define AVG_LOG 2.3978952727983707f /* log(11.0) */
#define PNA_EPS 1e-5f

__device__ __forceinline__ v8f wmma4(v2f a, v2f b, v8f c) {
  // D = A(16x4,f32) * B(4x16,f32) + C(16x16,f32)
  return __builtin_amdgcn_wmma_f32_16x16x4_f32(false, a, false, b, (short)0, c,
                                               false, false);
}

// float atomic min/max via order-preserving integer punning
__device__ __forceinline__ void atomicMinF(float* a, float v) {
  if (v >= 0.0f) atomicMin((int*)a, __float_as_int(v));
  else           atomicMax((unsigned int*)a, __float_as_uint(v));
}
__device__ __forceinline__ void atomicMaxF(float* a, float v) {
  if (v >= 0.0f) atomicMax((int*)a, __float_as_int(v));
  else           atomicMin((unsigned int*)a, __float_as_uint(v));
}

// ---------------------------------------------------------------------------
// Generic row-major GEMM: C[M,N] = A[M,K] @ W[K,N] (+bias) (+relu)
// One wave per 16x16 output tile. Requires M%16==0, N%16==0, K%4==0.
// A fragment (16x4): lane l -> row=l&15, K = 2*(l>>4) + {0,1}
// B fragment (4x16): lane l -> col=l&15, K = 2*(l>>4) + {0,1}
// C/D: VGPR r, lanes 0-15 -> row r, lanes 16-31 -> row r+8; col = l&15
// ---------------------------------------------------------------------------
__global__ void gemm_wmma_f32(const float* __restrict__ A, int lda,
                              const float* __restrict__ W, int ldw,
                              const float* __restrict__ bias,
                              float* __restrict__ C, int ldc,
                              int M, int N, int K, int do_relu) {
  int gwave = (blockIdx.x * blockDim.x + threadIdx.x) >> 5;
  int lane  = threadIdx.x & 31;
  int tn_cnt = N >> 4;
  int tiles  = (M >> 4) * tn_cnt;
  if (gwave >= tiles) return;
  int tm = gwave / tn_cnt, tn = gwave % tn_cnt;
  int r = lane & 15, g = lane >> 4;
  int row = tm * 16 + r;
  int col = tn * 16 + r;

  v8f acc = {};
  for (int k0 = 0; k0 < K; k0 += 4) {
    int ka = k0 + 2 * g;
    v2f a;
    a.x = A[(size_t)row * lda + ka];
    a.y = A[(size_t)row * lda + ka + 1];
    v2f b;
    b.x = W[(size_t)ka * ldw + col];
    b.y = W[(size_t)(ka + 1) * ldw + col];
    acc = wmma4(a, b, acc);
  }
  float bv = bias ? bias[col] : 0.0f;
#pragma unroll
  for (int rr = 0; rr < 8; ++rr) {
    int orow = tm * 16 + rr + 8 * g;
    float v = acc[rr] + bv;
    if (do_relu) v = fmaxf(v, 0.0f);
    C[(size_t)orow * ldc + col] = v;
  }
}

// ---------------------------------------------------------------------------
// Small utility kernels
// ---------------------------------------------------------------------------
__global__ void zero_f32(float* __restrict__ p, int n) {
  int i = blockIdx.x * blockDim.x + threadIdx.x;
  if (i < n) p[i] = 0.0f;
}

__global__ void copy_x_to_hcat(const float* __restrict__ x,
                               float* __restrict__ H, int n) {  // n = N*64
  int i = blockIdx.x * blockDim.x + threadIdx.x;
  if (i < n) {
    int row = i >> 6, f = i & 63;
    H[(size_t)row * 128 + f] = x[i];
  }
}

__global__ void count_deg(const int* __restrict__ ei, int E,
                          float* __restrict__ deg) {
  int e = blockIdx.x * blockDim.x + threadIdx.x;
  if (e < E) atomicAdd(&deg[ei[E + e]], 1.0f);  // dst = ei[1][e]
}

__global__ void finish_deg(float* __restrict__ degc, float* __restrict__ has,
                           float* __restrict__ amp, float* __restrict__ inv,
                           int n) {
  int i = blockIdx.x * blockDim.x + threadIdx.x;
  if (i < n) {
    float d = degc[i];
    float h = (d > 0.0f) ? 1.0f : 0.0f;
    float dc = fmaxf(d, 1.0f);
    float a = logf(dc + 1.0f) / AVG_LOG;
    degc[i] = dc;
    has[i]  = h;
    amp[i]  = a;
    inv[i]  = 1.0f / a;
  }
}

__global__ void init_agg(float* __restrict__ S, float* __restrict__ S2,
                         float* __restrict__ MN, float* __restrict__ MX,
                         int n) {
  int i = blockIdx.x * blockDim.x + threadIdx.x;
  if (i < n) {
    S[i] = 0.0f;
    S2[i] = 0.0f;
    MN[i] = __builtin_inff();
    MX[i] = -__builtin_inff();
  }
}

// per (edge, float4 feature group): m = P[dst] + Q[src]; scatter aggregates
__global__ void edge_agg(const int* __restrict__ ei, int E,
                         const float* __restrict__ P, const float* __restrict__ Q,
                         float* __restrict__ S, float* __restrict__ S2,
                         float* __restrict__ MN, float* __restrict__ MX) {
  int tid = blockIdx.x * blockDim.x + threadIdx.x;
  int e = tid >> 4;
  if (e >= E) return;
  int f = (tid & 15) << 2;
  int src = ei[e];
  int dst = ei[E + e];
  const float4 p = *(const float4*)(P + (size_t)dst * 64 + f);
  const float4 q = *(const float4*)(Q + (size_t)src * 64 + f);
  float m[4] = {p.x + q.x, p.y + q.y, p.z + q.z, p.w + q.w};
  size_t base = (size_t)dst * 64 + f;
#pragma unroll
  for (int i = 0; i < 4; ++i) {
    atomicAdd(&S[base + i], m[i]);
    atomicAdd(&S2[base + i], m[i] * m[i]);
    atomicMinF(&MN[base + i], m[i]);
    atomicMaxF(&MX[base + i], m[i]);
  }
}

// ---------------------------------------------------------------------------
// Fused PNA post stage, one wave per 16-node tile:
//   post = h@Wx + agg@Wid + amp.(agg@Wamp) + (1/amp).(agg@Watt) + post_b
//   out  = relu(post@lin_w + lin_b)     (written into hcat column slice)
// agg columns built on the fly from SUM/SSQ/MIN/MAX + deg stats.
// ---------------------------------------------------------------------------
__global__ void pna_post(const float* __restrict__ H, int ldh,
                         const float* __restrict__ S, const float* __restrict__ S2,
                         const float* __restrict__ MN, const float* __restrict__ MX,
                         const float* __restrict__ DEGC, const float* __restrict__ HAS,
                         const float* __restrict__ AMP, const float* __restrict__ INV,
                         const float* __restrict__ PW, const float* __restrict__ PB,
                         const float* __restrict__ LW, const float* __restrict__ LB,
                         float* __restrict__ OUT, int ldo, int mtiles) {
  __shared__ float pbuf[8][16][33];  // per-wave 16x32 post tile (+pad)
  int gwave = (blockIdx.x * blockDim.x + threadIdx.x) >> 5;
  if (gwave >= mtiles) return;
  int wib = (threadIdx.x >> 5) & 7;
  int lane = threadIdx.x & 31;
  int r = lane & 15, g = lane >> 4;
  int row = gwave * 16 + r;

  float inv_dc = 1.0f / DEGC[row];
  float hs = HAS[row];

  v8f aX0 = {}, aX1 = {}, aI0 = {}, aI1 = {}, aA0 = {}, aA1 = {}, aT0 = {}, aT1 = {};

  // h @ Wx  (K = 64; Wx = post_w rows [0,64))
  for (int k0 = 0; k0 < 64; k0 += 4) {
    int ka = k0 + 2 * g;
    v2f a;
    a.x = H[(size_t)row * ldh + ka];
    a.y = H[(size_t)row * ldh + ka + 1];
    v2f b0, b1;
    b0.x = PW[ka * 32 + r];      b0.y = PW[(ka + 1) * 32 + r];
    b1.x = PW[ka * 32 + r + 16]; b1.y = PW[(ka + 1) * 32 + r + 16];
    aX0 = wmma4(a, b0, aX0);
    aX1 = wmma4(a, b1, aX1);
  }

  const float* WI = PW + 64 * 32;   // identity-scaler rows  [64,384)
  const float* WA = PW + 384 * 32;  // amplification rows    [384,704)
  const float* WT = PW + 704 * 32;  // attenuation rows      [704,1024)

  // agg @ {Wid,Wamp,Watt}  (K = 320 = 5 aggregators x 64)
  for (int k0 = 0; k0 < 320; k0 += 4) {
    int ka = k0 + 2 * g;
    int seg = ka >> 6;          // wave-uniform
    int f = ka & 63;
    size_t base = (size_t)row * 64 + f;
    float a0, a1;
    if (seg == 0) {             // mean
      a0 = S[base] * inv_dc;  a1 = S[base + 1] * inv_dc;
    } else if (seg == 1) {      // sum
      a0 = S[base];           a1 = S[base + 1];
    } else if (seg == 2) {      // std
      float m0 = S[base] * inv_dc, m1 = S[base + 1] * inv_dc;
      float q0 = S2[base] * inv_dc, q1 = S2[base + 1] * inv_dc;
      a0 = sqrtf(fmaxf(q0 - m0 * m0, 0.0f) + PNA_EPS);
      a1 = sqrtf(fmaxf(q1 - m1 * m1, 0.0f) + PNA_EPS);
    } else if (seg == 3) {      // min (masked if deg==0)
      a0 = (hs > 0.5f) ? MN[base] : 0.0f;
      a1 = (hs > 0.5f) ? MN[base + 1] : 0.0f;
    } else {                    // max
      a0 = (hs > 0.5f) ? MX[base] : 0.0f;
      a1 = (hs > 0.5f) ? MX[base + 1] : 0.0f;
    }
    v2f a; a.x = a0; a.y = a1;
    v2f b;
    b.x = WI[ka * 32 + r];      b.y = WI[(ka + 1) * 32 + r];      aI0 = wmma4(a, b, aI0);
    b.x = WI[ka * 32 + r + 16]; b.y = WI[(ka + 1) * 32 + r + 16]; aI1 = wmma4(a, b, aI1);
    b.x = WA[ka * 32 + r];      b.y = WA[(ka + 1) * 32 + r];      aA0 = wmma4(a, b, aA0);
    b.x = WA[ka * 32 + r + 16]; b.y = WA[(ka + 1) * 32 + r + 16]; aA1 = wmma4(a, b, aA1);
    b.x = WT[ka * 32 + r];      b.y = WT[(ka + 1) * 32 + r];      aT0 = wmma4(a, b, aT0);
    b.x = WT[ka * 32 + r + 16]; b.y = WT[(ka + 1) * 32 + r + 16]; aT1 = wmma4(a, b, aT1);
  }

  // combine scaler streams (row scalars commute with GEMM) -> LDS stage
  float pb0 = PB[r], pb1 = PB[r + 16];
#pragma unroll
  for (int rr = 0; rr < 8; ++rr) {
    int orow = gwave * 16 + rr + 8 * g;
    float am = AMP[orow], iv = INV[orow];
    pbuf[wib][rr + 8 * g][r]      = aX0[rr] + aI0[rr] + am * aA0[rr] + iv * aT0[rr] + pb0;
    pbuf[wib][rr + 8 * g][r + 16] = aX1[rr] + aI1[rr] + am * aA1[rr] + iv * aT1[rr] + pb1;
  }
  __threadfence_block();  // ordering of same-wave LDS RAW (DS ops in-order)

  // lin: out = relu(post @ LW + LB), K = 32
  v8f o0 = {}, o1 = {};
  for (int k0 = 0; k0 < 32; k0 += 4) {
    int ka = k0 + 2 * g;
    v2f a;
    a.x = pbuf[wib][r][ka];
    a.y = pbuf[wib][r][ka + 1];
    v2f b0, b1;
    b0.x = LW[ka * 32 + r];      b0.y = LW[(ka + 1) * 32 + r];
    b1.x = LW[ka * 32 + r + 16]; b1.y = LW[(ka + 1) * 32 + r + 16];
    o0 = wmma4(a, b0, o0);
    o1 = wmma4(a, b1, o1);
  }
  float lb0 = LB[r], lb1 = LB[r + 16];
#pragma unroll
  for (int rr = 0; rr < 8; ++rr) {
    int orow = gwave * 16 + rr + 8 * g;
    OUT[(size_t)orow * ldo + r]      = fmaxf(o0[rr] + lb0, 0.0f);
    OUT[(size_t)orow * ldo + r + 16] = fmaxf(o1[rr] + lb1, 0.0f);
  }
}

// final head: out[i] = h[i,:64] . out_w + out_b
__global__ void head_dot(const float* __restrict__ H, const float* __restrict__ w,
                         const float* __restrict__ b, float* __restrict__ out,
                         int n) {
  int i = blockIdx.x * blockDim.x + threadIdx.x;
  if (i < n) {
    const float* h = H + (size_t)i * 128;
    float acc = b[0];
#pragma unroll
    for (int f = 0; f < 64; ++f) acc += h[f] * w[f];
    out[i] = acc;
  }
}

// ---------------------------------------------------------------------------
static inline int cdiv(int a, int b) { return (a + b - 1) / b; }

static void launch_gemm(const float* A, int lda, const float* W, int ldw,
                        const float* bias, float* C, int ldc, int M, int N,
                        int K, int relu, hipStream_t s) {
  int tiles = (M / 16) * (N / 16);
  int blocks = cdiv(tiles * 32, 256);
  gemm_wmma_f32<<<blocks, 256, 0, s>>>(A, lda, W, ldw, bias, C, ldc, M, N, K, relu);
}

extern "C" void kernel_launch(void* const* d_in, const int* in_sizes, int n_in,
                              void* d_out, int out_size, void* d_ws,
                              size_t ws_size, hipStream_t stream) {
  const float* x        = (const float*)d_in[0];
  const int*   ei_in    = (const int*)d_in[1];
  const int*   ei_out   = (const int*)d_in[2];
  const float* pre_w[2]  = {(const float*)d_in[3], (const float*)d_in[9]};
  const float* pre_b[2]  = {(const float*)d_in[4], (const float*)d_in[10]};
  const float* post_w[2] = {(const float*)d_in[5], (const float*)d_in[11]};
  const float* post_b[2] = {(const float*)d_in[6], (const float*)d_in[12]};
  const float* lin_w[2]  = {(const float*)d_in[7], (const float*)d_in[13]};
  const float* lin_b[2]  = {(const float*)d_in[8], (const float*)d_in[14]};
  const float* comb_w   = (const float*)d_in[15];
  const float* comb_b   = (const float*)d_in[16];
  const float* out_w    = (const float*)d_in[17];
  const float* out_b    = (const float*)d_in[18];
  const int*   ei[2]    = {ei_in, ei_out};

  int Nn = in_sizes[0] / 64;  // 50000 (multiple of 16)
  int E  = in_sizes[1] / 2;   // 500000

  // workspace carve-up (~130 MB of f32, L2-resident hot set)
  float* ws = (float*)d_ws;
  size_t off = 0;
  auto carve = [&](size_t cnt) { float* p = ws + off; off += cnt; return p; };
  float* HA  = carve((size_t)Nn * 128);  // [h | h_in | h_out] layer-even
  float* HB  = carve((size_t)Nn * 128);  // layer-odd
  float* P   = carve((size_t)Nn * 64);   // h @ W_top   (dst side)
  float* Q   = carve((size_t)Nn * 64);   // h @ W_bot+b (src side)
  float* S   = carve((size_t)Nn * 64);
  float* S2  = carve((size_t)Nn * 64);
  float* MNb = carve((size_t)Nn * 64);
  float* MXb = carve((size_t)Nn * 64);
  float* degc[2], *has[2], *amp[2], *inv[2];
  for (int d = 0; d < 2; ++d) {
    degc[d] = carve(Nn); has[d] = carve(Nn); amp[d] = carve(Nn); inv[d] = carve(Nn);
  }

  const int T = 256;

  // stage h=x into the 128-wide concat buffer
  copy_x_to_hcat<<<cdiv(Nn * 64, T), T, 0, stream>>>(x, HA, Nn * 64);

  // degrees + scalers per direction (edge lists fixed across layers)
  for (int d = 0; d < 2; ++d) {
    zero_f32<<<cdiv(Nn, T), T, 0, stream>>>(degc[d], Nn);
    count_deg<<<cdiv(E, T), T, 0, stream>>>(ei[d], E, degc[d]);
    finish_deg<<<cdiv(Nn, T), T, 0, stream>>>(degc[d], has[d], amp[d], inv[d], Nn);
  }

  float* cur = HA;
  float* nxt = HB;
  for (int l = 0; l < 2; ++l) {
    for (int d = 0; d < 2; ++d) {
      const float* pw = pre_w[d] + (size_t)l * 128 * 64;
      // edge MLP factored into two node GEMMs: m_e = P[dst] + Q[src]
      launch_gemm(cur, 128, pw, 64, nullptr, P, 64, Nn, 64, 64, 0, stream);
      launch_gemm(cur, 128, pw + 64 * 64, 64, pre_b[d] + (size_t)l * 64, Q, 64,
                  Nn, 64, 64, 0, stream);
      init_agg<<<cdiv(Nn * 64, T), T, 0, stream>>>(S, S2, MNb, MXb, Nn * 64);
      edge_agg<<<cdiv(E * 16, T), T, 0, stream>>>(ei[d], E, P, Q, S, S2, MNb, MXb);
      // fused post_nn + lin (+relu) -> hcat column slice [64,96) or [96,128)
      int mtiles = Nn / 16;
      pna_post<<<cdiv(mtiles * 32, T), T, 0, stream>>>(
          cur, 128, S, S2, MNb, MXb, degc[d], has[d], amp[d], inv[d],
          post_w[d] + (size_t)l * 1024 * 32, post_b[d] + (size_t)l * 32,
          lin_w[d] + (size_t)l * 32 * 32, lin_b[d] + (size_t)l * 32,
          cur + 64 + 32 * d, 128, mtiles);
    }
    // comb: hnext = relu([h|h_in|h_out] @ comb_w + comb_b)
    launch_gemm(cur, 128, comb_w + (size_t)l * 128 * 64, 64,
                comb_b + (size_t)l * 64, nxt, 128, Nn, 64, 128, 1, stream);
    float* t = cur; cur = nxt; nxt = t;
  }

  head_dot<<<cdiv(Nn, T), T, 0, stream>>>(cur, out_w, out_b, (float*)d_out, Nn);
}